// ENNModelWithSparsityControl_3324304687192
// MI455X (gfx1250) — compile-verified
//
#include <hip/hip_runtime.h>

typedef __attribute__((ext_vector_type(2))) float v2f;
typedef __attribute__((ext_vector_type(8))) float v8f;

namespace {
constexpr int kBatch = 2048;
constexpr int kN = 256;
constexpr int kS = 32;
constexpr int kBT = 32;  // batch tile per workgroup
constexpr int kNT = 32;  // n tile per workgroup
}  // namespace

// ---------------------------------------------------------------------------
// Kernel A: B0[s][k][m] = sigmoid(ent[k][s]) * mixing[k][m]   (32x256x256, 8MB)
// ---------------------------------------------------------------------------
__global__ void enn_build_b0(const float* __restrict__ ent,
                             const float* __restrict__ mixing,
                             float* __restrict__ B0) {
  const int tid = blockIdx.x * blockDim.x + threadIdx.x;  // 0 .. 32*256*256-1
  const int m = tid & (kN - 1);
  const int k = (tid >> 8) & (kN - 1);
  const int s = tid >> 16;
  const float e = ent[k * kS + s];
  const float sig = 1.0f / (1.0f + __expf(-e));
  B0[tid] = sig * mixing[k * kN + m];
}

// ---------------------------------------------------------------------------
// Kernel B: per-s matrix square  C[s] = A[s] @ A[s]  (32 slices of 256x256)
// One wave computes one 16x16 output tile via V_WMMA_F32_16X16X4_F32.
// kTransposeOut=true writes C transposed ([s][n][k]) so the consumer can load
// B-fragments as contiguous float2 (and the 8 D values per lane coalesce into
// contiguous b128-mergeable stores).
// ---------------------------------------------------------------------------
template <bool kTransposeOut>
__global__ void __launch_bounds__(256) enn_matsq(const float* __restrict__ A,
                                                 float* __restrict__ C) {
  const int wave = (blockIdx.x * blockDim.x + threadIdx.x) >> 5;
  const int lane = threadIdx.x & 31;
  const int lm = lane & 15;
  const int half = lane >> 4;       // lane half selects K+2 per ISA A/B layout
  const int tj = wave & 15;         // N tile
  const int ti = (wave >> 4) & 15;  // M tile
  const int s = wave >> 8;          // slice
  const float* __restrict__ As = A + (size_t)s * kN * kN;
  float* __restrict__ Cs = C + (size_t)s * kN * kN;
  const int row = ti * 16 + lm;     // A fragment: M = lane%16
  const int col = tj * 16 + lm;     // B fragment: N = lane%16

  v8f acc;
#pragma unroll
  for (int r = 0; r < 8; ++r) acc[r] = 0.0f;

  for (int k = 0; k < kN; k += 4) {
    const int ka = k + half * 2;
    v2f a = *(const v2f*)(As + (size_t)row * kN + ka);  // A[row][ka..ka+1]
    v2f b;
    b.x = As[(size_t)(ka + 0) * kN + col];              // B[ka+0][col]
    b.y = As[(size_t)(ka + 1) * kN + col];              // B[ka+1][col]
    acc = __builtin_amdgcn_wmma_f32_16x16x4_f32(false, a, false, b, (short)0,
                                                acc, false, false);
  }
  // D layout: VGPR r holds M = r + 8*half, N = lane%16
  if (kTransposeOut) {
    float* __restrict__ dst = Cs + (size_t)col * kN + ti * 16 + 8 * half;
#pragma unroll
    for (int r = 0; r < 8; ++r) dst[r] = acc[r];  // contiguous per lane
  } else {
#pragma unroll
    for (int r = 0; r < 8; ++r) {
      Cs[(size_t)(ti * 16 + r + 8 * half) * kN + col] = acc[r];
    }
  }
}

// ---------------------------------------------------------------------------
// Kernel C (fused main):
//   tmp[b][n][s] = sum_k x[b][k][s] * A4[s][k][n]      (per-s GEMM, WMMA)
//   out[b][n][t] = sum_s tmp[b][n][s] * W[t][s]        (readout, WMMA)
// Workgroup = 32 b-rows x 32 n-cols, 8 waves; wave wv owns s = 4*wv..4*wv+3
// with a 2x2 grid of 16x16 accumulator tiles per s (128 acc VGPRs/wave).
// A4 arrives transposed: A4t[s][n][k] -> B fragments are contiguous float2.
// 64 KB LDS, reused: x chunk stage [32][16][32], then tmp in two bsub passes.
// ---------------------------------------------------------------------------
__global__ void __launch_bounds__(256)
enn_main(const float* __restrict__ x, const float* __restrict__ A4t,
         const float* __restrict__ W, float* __restrict__ out) {
  __shared__ float lds[16384];  // 64 KB

  const int bt = blockIdx.x >> 3;  // 0..63  batch tile
  const int nt = blockIdx.x & 7;   // 0..7   n tile
  const int tid = threadIdx.x;
  const int wv = tid >> 5;         // wave in block, 0..7
  const int lane = tid & 31;
  const int lm = lane & 15;
  const int half = lane >> 4;

  const int b0 = bt * kBT;
  const int n0 = nt * kNT;

  v8f acc[4][2][2];  // [ss][bsub][nsub]
#pragma unroll
  for (int i = 0; i < 4; ++i)
#pragma unroll
    for (int p = 0; p < 2; ++p)
#pragma unroll
      for (int q = 0; q < 2; ++q)
#pragma unroll
        for (int r = 0; r < 8; ++r) acc[i][p][q][r] = 0.0f;

  for (int kc = 0; kc < kN; kc += 16) {
    // Stage x[b0..b0+31, kc..kc+15, 0..31] -> lds[b][kk][s] (32*512 floats).
    // Per b-row this is 512 consecutive floats in global memory.
#pragma unroll
    for (int i = 0; i < 16; ++i) {
      const int f = (tid + i * 256) * 4;  // float index, float4 aligned
      const int b = f >> 9;               // /512
      const int off = f & 511;
      *(float4*)(lds + f) =
          *(const float4*)(x + (size_t)(b0 + b) * (kN * kS) + kc * kS + off);
    }
    __syncthreads();

#pragma unroll
    for (int ss = 0; ss < 4; ++ss) {
      const int s = wv * 4 + ss;
#pragma unroll
      for (int k4 = 0; k4 < 4; ++k4) {
        const int kl = k4 * 4 + half * 2;
        const int ka = kc + kl;
        v2f a[2];  // A fragments: X[b][ka..ka+1][s], bsub = 0,1
#pragma unroll
        for (int p = 0; p < 2; ++p) {
          a[p].x = lds[(p * 16 + lm) * 512 + kl * 32 + s];
          a[p].y = lds[(p * 16 + lm) * 512 + (kl + 1) * 32 + s];
        }
        v2f b[2];  // B fragments: A4t[s][col][ka..ka+1], contiguous float2
#pragma unroll
        for (int q = 0; q < 2; ++q) {
          const int col = n0 + q * 16 + lm;
          b[q] = *(const v2f*)(A4t + ((size_t)s * kN + col) * kN + ka);
        }
#pragma unroll
        for (int p = 0; p < 2; ++p)
#pragma unroll
          for (int q = 0; q < 2; ++q)
            acc[ss][p][q] = __builtin_amdgcn_wmma_f32_16x16x4_f32(
                false, a[p], false, b[q], (short)0, acc[ss][p][q], false,
                false);
      }
    }
    __syncthreads();
  }

  // Readout in two passes (bsub = 0,1), each using 64 KB tmp[16][32][32].
#pragma unroll 1
  for (int bsub = 0; bsub < 2; ++bsub) {
    // Spill tmp[bl][n][s] for this bsub.
#pragma unroll
    for (int ss = 0; ss < 4; ++ss) {
      const int s = wv * 4 + ss;
#pragma unroll
      for (int q = 0; q < 2; ++q)
#pragma unroll
        for (int r = 0; r < 8; ++r) {
          const int bl = r + 8 * half;
          const int n = q * 16 + lm;
          lds[bl * 1024 + n * 32 + s] = acc[ss][bsub][q][r];
        }
    }
    __syncthreads();

    // out[b][n][t] = sum_s tmp[bl][n][s] * W[t][s]
    // 64 tiles (32 n-locals x 2 t-halves), 8 per wave, K = 32.
#pragma unroll
    for (int j = 0; j < 8; ++j) {
      const int tile = wv * 8 + j;
      const int nl = tile >> 1;  // n-local 0..31
      const int th = tile & 1;   // t half
      v8f d;
#pragma unroll
      for (int r = 0; r < 8; ++r) d[r] = 0.0f;
#pragma unroll
      for (int k = 0; k < kS; k += 4) {
        const int ka = k + half * 2;
        v2f a = *(const v2f*)(lds + lm * 1024 + nl * 32 + ka);  // tmp frag
        v2f b = *(const v2f*)(W + (th * 16 + lm) * kS + ka);    // W[t][ka..+1]
        d = __builtin_amdgcn_wmma_f32_16x16x4_f32(false, a, false, b, (short)0,
                                                  d, false, false);
      }
#pragma unroll
      for (int r = 0; r < 8; ++r) {
        const int bb = b0 + bsub * 16 + r + 8 * half;
        const int nn = n0 + nl;
        const int tt = th * 16 + lm;
        out[((size_t)bb * kN + nn) * kS + tt] = d[r];
      }
    }
    __syncthreads();  // tmp fully consumed before next bsub overwrites it
  }
}

// ---------------------------------------------------------------------------
// Launch: B0 -> B2 -> A4^T (=B0^4 transposed) -> fused main pass.
// Workspace: 2 x 8 MB ping-pong buffers.
// ---------------------------------------------------------------------------
extern "C" void kernel_launch(void* const* d_in, const int* in_sizes, int n_in,
                              void* d_out, int out_size, void* d_ws,
                              size_t ws_size, hipStream_t stream) {
  const float* x = (const float*)d_in[0];
  const float* ent = (const float*)d_in[1];
  const float* mixing = (const float*)d_in[2];
  const float* readout_w = (const float*)d_in[3];
  float* out = (float*)d_out;

  float* ws0 = (float*)d_ws;                // 32*256*256 floats = 8 MB
  float* ws1 = ws0 + (size_t)kS * kN * kN;  // second 8 MB

  enn_build_b0<<<(kS * kN * kN) / 256, 256, 0, stream>>>(ent, mixing, ws0);
  // 32 s * 256 tiles = 8192 waves / 8 waves-per-block = 1024 blocks
  enn_matsq<false><<<(kS * 256) / 8, 256, 0, stream>>>(ws0, ws1);  // B2 = B0^2
  enn_matsq<true><<<(kS * 256) / 8, 256, 0, stream>>>(ws1, ws0);   // A4t
  enn_main<<<(kBatch / kBT) * (kN / kNT), 256, 0, stream>>>(x, ws0, readout_w,
                                                            out);
}